// _Image2Text_87531433493163
// MI455X (gfx1250) — compile-verified
//
#include <hip/hip_runtime.h>
#include <math.h>

#define NB 8          // batch
#define NK 4          // beams
#define NBK 32        // NB*NK
#define DD 1024       // hidden
#define VV 32000      // vocab
#define TT 16         // steps
#define FF 4096       // 4*DD
#define EOS_TOK 2
#define PAD_TOK 0
#define EOS_SCORE 1000.0f
#define BIGF 1e9f

typedef __attribute__((ext_vector_type(2))) float v2f;
typedef __attribute__((ext_vector_type(8))) float v8f;

__device__ __forceinline__ v8f wmma4(v2f a, v2f b, v8f c) {
  // D = A(16x4,f32) * B(4x16,f32) + C(16x16,f32)
  return __builtin_amdgcn_wmma_f32_16x16x4_f32(false, a, false, b, (short)0, c, false, false);
}

__device__ __forceinline__ float sigm(float x) { return 1.0f / (1.0f + expf(-x)); }

// ---------------- init ----------------
__global__ void k_init(const int* __restrict__ words, const float* __restrict__ h0,
                       const float* __restrict__ c0, int* __restrict__ w,
                       float* __restrict__ scores, int* __restrict__ stopped,
                       float* __restrict__ h_cur, float* __restrict__ c_cur,
                       int* __restrict__ bufA, int* __restrict__ bufB) {
  int idx = blockIdx.x * blockDim.x + threadIdx.x;
  if (idx < NBK * DD) { h_cur[idx] = h0[idx]; c_cur[idx] = c0[idx]; }
  if (idx < NBK * TT) { bufA[idx] = 0; bufB[idx] = 0; }
  if (idx < NBK) {
    w[idx] = words[idx];
    scores[idx] = ((idx & (NK - 1)) == 0) ? 0.0f : BIGF;
  }
  if (idx < NB) stopped[idx] = 0;
}

// ---------------- gates GEMM + LSTM cell (fused) ----------------
// grid: DD/16 blocks of 32 threads; each wave computes BOTH 16-row M-tiles for
// its 16 d-columns (4 gate tiles x 2 m-tiles = 8 WMMA accumulators), so every
// weight (B) load feeds two WMMAs.
__global__ void k_gates_lstm(const float* __restrict__ E, const float* __restrict__ Wx,
                             const float* __restrict__ Wh, const float* __restrict__ bias,
                             const int* __restrict__ w, const float* __restrict__ h_cur,
                             const float* __restrict__ c_cur,
                             float* __restrict__ h_tmp, float* __restrict__ c_tmp) {
  const int lane = threadIdx.x;
  const int n0 = blockIdx.x << 4;      // d offset
  const int nl = lane & 15;
  const int khalf = (lane >> 4) << 1;  // 0 or 2
  const float* xrow0 = E + (size_t)w[nl] * DD;        // m-tile 0 rows 0..15
  const float* xrow1 = E + (size_t)w[16 + nl] * DD;   // m-tile 1 rows 16..31
  const float* hrow0 = h_cur + (size_t)nl * DD;
  const float* hrow1 = h_cur + (size_t)(16 + nl) * DD;
  const int c0 = n0 + nl;

  v8f acc[2][4];
#pragma unroll
  for (int mt = 0; mt < 2; ++mt)
#pragma unroll
    for (int g = 0; g < 4; ++g) acc[mt][g] = (v8f){};

  // phase 1: x @ Wx
  for (int kk = 0; kk < DD; kk += 4) {
    const int kb = kk + khalf;
    v2f a0; a0.x = xrow0[kb]; a0.y = xrow0[kb + 1];
    v2f a1; a1.x = xrow1[kb]; a1.y = xrow1[kb + 1];
    const float* Bp = Wx + (size_t)kb * FF;
    v2f b0, b1, b2, b3;
    b0.x = Bp[c0];          b0.y = Bp[FF + c0];
    b1.x = Bp[DD + c0];     b1.y = Bp[FF + DD + c0];
    b2.x = Bp[2 * DD + c0]; b2.y = Bp[FF + 2 * DD + c0];
    b3.x = Bp[3 * DD + c0]; b3.y = Bp[FF + 3 * DD + c0];
    acc[0][0] = wmma4(a0, b0, acc[0][0]);
    acc[1][0] = wmma4(a1, b0, acc[1][0]);
    acc[0][1] = wmma4(a0, b1, acc[0][1]);
    acc[1][1] = wmma4(a1, b1, acc[1][1]);
    acc[0][2] = wmma4(a0, b2, acc[0][2]);
    acc[1][2] = wmma4(a1, b2, acc[1][2]);
    acc[0][3] = wmma4(a0, b3, acc[0][3]);
    acc[1][3] = wmma4(a1, b3, acc[1][3]);
  }
  // phase 2: h @ Wh
  for (int kk = 0; kk < DD; kk += 4) {
    const int kb = kk + khalf;
    v2f a0; a0.x = hrow0[kb]; a0.y = hrow0[kb + 1];
    v2f a1; a1.x = hrow1[kb]; a1.y = hrow1[kb + 1];
    const float* Bp = Wh + (size_t)kb * FF;
    v2f b0, b1, b2, b3;
    b0.x = Bp[c0];          b0.y = Bp[FF + c0];
    b1.x = Bp[DD + c0];     b1.y = Bp[FF + DD + c0];
    b2.x = Bp[2 * DD + c0]; b2.y = Bp[FF + 2 * DD + c0];
    b3.x = Bp[3 * DD + c0]; b3.y = Bp[FF + 3 * DD + c0];
    acc[0][0] = wmma4(a0, b0, acc[0][0]);
    acc[1][0] = wmma4(a1, b0, acc[1][0]);
    acc[0][1] = wmma4(a0, b1, acc[0][1]);
    acc[1][1] = wmma4(a1, b1, acc[1][1]);
    acc[0][2] = wmma4(a0, b2, acc[0][2]);
    acc[1][2] = wmma4(a1, b2, acc[1][2]);
    acc[0][3] = wmma4(a0, b3, acc[0][3]);
    acc[1][3] = wmma4(a1, b3, acc[1][3]);
  }

  const int d = n0 + nl;
  const float bi_ = bias[d];
  const float bf_ = bias[DD + d];
  const float bg_ = bias[2 * DD + d];
  const float bo_ = bias[3 * DD + d];
#pragma unroll
  for (int mt = 0; mt < 2; ++mt) {
    const int mbase = (mt << 4) + ((lane >> 4) << 3);
#pragma unroll
    for (int r = 0; r < 8; ++r) {
      const int row = mbase + r;
      float iv = acc[mt][0][r] + bi_;
      float fv = acc[mt][1][r] + bf_;
      float gv = acc[mt][2][r] + bg_;
      float ov = acc[mt][3][r] + bo_;
      float co = c_cur[(size_t)row * DD + d];
      float cn = sigm(fv) * co + sigm(iv) * tanhf(gv);
      float hn = sigm(ov) * tanhf(cn);
      c_tmp[(size_t)row * DD + d] = cn;
      h_tmp[(size_t)row * DD + d] = hn;
    }
  }
}

// ---------------- logits GEMM ----------------
// grid: VV/16 blocks of 32 threads; each wave computes both M-tiles so every
// Wout load feeds two WMMAs (weight traffic is the roofline bottleneck).
__global__ void k_logits(const float* __restrict__ h_tmp, const float* __restrict__ Wout,
                         const float* __restrict__ bo, float* __restrict__ logits) {
  const int lane = threadIdx.x;
  const int n0 = blockIdx.x << 4;
  const int nl = lane & 15;
  const int khalf = (lane >> 4) << 1;
  const float* arow0 = h_tmp + (size_t)nl * DD;
  const float* arow1 = h_tmp + (size_t)(16 + nl) * DD;
  const int col = n0 + nl;
  v8f acc0 = {}, acc1 = {};
  for (int kk = 0; kk < DD; kk += 4) {
    const int kb = kk + khalf;
    v2f a0; a0.x = arow0[kb]; a0.y = arow0[kb + 1];
    v2f a1; a1.x = arow1[kb]; a1.y = arow1[kb + 1];
    const float* Bp = Wout + (size_t)kb * VV + col;
    v2f bv; bv.x = Bp[0]; bv.y = Bp[VV];
    acc0 = wmma4(a0, bv, acc0);
    acc1 = wmma4(a1, bv, acc1);
  }
  const float bb = bo[col];
  const int msub = (lane >> 4) << 3;
#pragma unroll
  for (int r = 0; r < 8; ++r)
    logits[(size_t)(msub + r) * VV + col] = acc0[r] + bb;
#pragma unroll
  for (int r = 0; r < 8; ++r)
    logits[(size_t)(16 + msub + r) * VV + col] = acc1[r] + bb;
}

// ---------------- per-row logsumexp + top-4 ----------------
__device__ __forceinline__ void top4_insert(float* bv, int* bi, float x, int idx) {
#pragma unroll
  for (int q = 0; q < 4; ++q) {
    if ((x > bv[q]) || (x == bv[q] && idx < bi[q])) {
      for (int r = 3; r > q; --r) { bv[r] = bv[r - 1]; bi[r] = bi[r - 1]; }
      bv[q] = x; bi[q] = idx;
      return;
    }
  }
}

__global__ void k_rowtopk(const float* __restrict__ logits, float* __restrict__ lse,
                          float* __restrict__ topv, int* __restrict__ topi) {
  const int row = blockIdx.x;       // 0..31
  const int t = threadIdx.x;        // 0..255
  const float* L = logits + (size_t)row * VV;
  float m = -INFINITY, s = 0.0f;
  float bv[4] = {-INFINITY, -INFINITY, -INFINITY, -INFINITY};
  int bi[4] = {VV, VV, VV, VV};
  for (int idx = t; idx < VV; idx += 256) {
    float x = L[idx];
    if (x > m) { s = s * expf(m - x) + 1.0f; m = x; }
    else       { s += expf(x - m); }
    top4_insert(bv, bi, x, idx);
  }
  __shared__ float sm[256], ss[256], svv[1024];
  __shared__ int sii[1024];
  sm[t] = m; ss[t] = s;
#pragma unroll
  for (int q = 0; q < 4; ++q) { svv[t * 4 + q] = bv[q]; sii[t * 4 + q] = bi[q]; }
  __syncthreads();
  for (int off = 128; off > 0; off >>= 1) {
    if (t < off) {
      float mA = sm[t], sA = ss[t], mB = sm[t + off], sB = ss[t + off];
      float mN = fmaxf(mA, mB);
      sm[t] = mN;
      ss[t] = sA * expf(mA - mN) + sB * expf(mB - mN);
      float lv[4]; int li[4];
#pragma unroll
      for (int q = 0; q < 4; ++q) { lv[q] = svv[t * 4 + q]; li[q] = sii[t * 4 + q]; }
#pragma unroll
      for (int q = 0; q < 4; ++q) top4_insert(lv, li, svv[(t + off) * 4 + q], sii[(t + off) * 4 + q]);
#pragma unroll
      for (int q = 0; q < 4; ++q) { svv[t * 4 + q] = lv[q]; sii[t * 4 + q] = li[q]; }
    }
    __syncthreads();
  }
  if (t == 0) {
    lse[row] = sm[0] + logf(ss[0]);
#pragma unroll
    for (int q = 0; q < 4; ++q) { topv[row * 4 + q] = svv[q]; topi[row * 4 + q] = sii[q]; }
  }
}

// ---------------- candidate combine + EOS logic + buf reorder ----------------
__global__ void k_combine(const float* __restrict__ topv, const int* __restrict__ topi,
                          const float* __restrict__ lse, float* __restrict__ scores,
                          int* __restrict__ w, int* __restrict__ stopped,
                          const int* __restrict__ buf_in, int* __restrict__ buf_out,
                          int* __restrict__ prev_out, int j) {
  const int b = threadIdx.x;
  if (b >= NB) return;
  float cand[16]; int word[16];
#pragma unroll
  for (int k = 0; k < NK; ++k) {
    const int rk = b * NK + k;
    const float sc = scores[rk], ls = lse[rk];
#pragma unroll
    for (int q = 0; q < NK; ++q) {
      cand[k * NK + q] = sc + (ls - topv[rk * 4 + q]);   // -logsoftmax + score
      word[k * NK + q] = topi[rk * 4 + q];
    }
  }
  // 4 smallest candidates, ascending, lower flat index wins ties
  int sel[4]; float selv[4]; bool used[16];
#pragma unroll
  for (int q = 0; q < 16; ++q) used[q] = false;
  for (int s = 0; s < NK; ++s) {
    int best = -1; float bestv = INFINITY;
    for (int q = 0; q < 16; ++q)
      if (!used[q] && cand[q] < bestv) { bestv = cand[q]; best = q; }
    used[best] = true; sel[s] = best; selv[s] = bestv;
  }
  const int st = stopped[b];
  int prv[4], prevb[4], tok[4], nw[4];
  float ns[4];
#pragma unroll
  for (int s = 0; s < NK; ++s) { prv[s] = sel[s] / NK; nw[s] = word[sel[s]]; }
  const bool first_eos = (nw[0] == EOS_TOK) && !st;
#pragma unroll
  for (int s = 0; s < NK; ++s) {
    const bool eos = (nw[s] == EOS_TOK);
    const bool pen = eos && !first_eos && !st;
    float scv = pen ? EOS_SCORE : selv[s];
    int tk = first_eos ? EOS_TOK : nw[s];
    tok[s] = st ? PAD_TOK : tk;
    prevb[s] = st ? s : prv[s];
    ns[s] = st ? scores[b * NK + s] : scv;
  }
  // reorder beam histories, then write step-j token
  for (int k = 0; k < NK; ++k)
    for (int t = 0; t < TT; ++t)
      buf_out[(b * NK + k) * TT + t] = buf_in[(b * NK + prevb[k]) * TT + t];
#pragma unroll
  for (int k = 0; k < NK; ++k) buf_out[(b * NK + k) * TT + j] = tok[k];
#pragma unroll
  for (int k = 0; k < NK; ++k) {
    scores[b * NK + k] = ns[k];
    w[b * NK + k] = st ? w[b * NK + k] : nw[k];
    prev_out[b * NK + k] = prv[k];
  }
  stopped[b] = st | (first_eos ? 1 : 0);
}

// ---------------- gather h/c by originating beam ----------------
__global__ void k_gather(const float* __restrict__ h_tmp, const float* __restrict__ c_tmp,
                         const int* __restrict__ prev, float* __restrict__ h_cur,
                         float* __restrict__ c_cur) {
  const int bk = blockIdx.x;            // 0..31
  const int b = bk / NK;
  const int p = prev[bk];
  const float* hs = h_tmp + (size_t)(b * NK + p) * DD;
  const float* cs = c_tmp + (size_t)(b * NK + p) * DD;
  for (int d = threadIdx.x; d < DD; d += blockDim.x) {
    h_cur[(size_t)bk * DD + d] = hs[d];
    c_cur[(size_t)bk * DD + d] = cs[d];
  }
}

// ---------------- finalize: (buf as float, scores) ----------------
__global__ void k_final(const int* __restrict__ buf, const float* __restrict__ scores,
                        float* __restrict__ out) {
  int idx = blockIdx.x * blockDim.x + threadIdx.x;
  if (idx < NBK * TT) out[idx] = (float)buf[idx];
  else if (idx < NBK * TT + NBK) out[idx] = scores[idx - NBK * TT];
}

extern "C" void kernel_launch(void* const* d_in, const int* in_sizes, int n_in,
                              void* d_out, int out_size, void* d_ws, size_t ws_size,
                              hipStream_t stream) {
  (void)in_sizes; (void)n_in; (void)out_size; (void)ws_size;
  const int*   words = (const int*)d_in[0];
  const float* h0    = (const float*)d_in[1];
  const float* c0    = (const float*)d_in[2];
  const float* E     = (const float*)d_in[3];
  const float* Wx    = (const float*)d_in[4];
  const float* Wh    = (const float*)d_in[5];
  const float* bias  = (const float*)d_in[6];
  const float* Wout  = (const float*)d_in[7];
  const float* bout  = (const float*)d_in[8];

  char* p = (char*)d_ws;
  float* h_cur  = (float*)p; p += (size_t)NBK * DD * 4;
  float* c_cur  = (float*)p; p += (size_t)NBK * DD * 4;
  float* h_tmp  = (float*)p; p += (size_t)NBK * DD * 4;
  float* c_tmp  = (float*)p; p += (size_t)NBK * DD * 4;
  float* logits = (float*)p; p += (size_t)NBK * VV * 4;
  float* lse    = (float*)p; p += 128;
  float* topv   = (float*)p; p += NBK * 4 * 4;
  int*   topi   = (int*)p;   p += NBK * 4 * 4;
  int*   wbuf   = (int*)p;   p += 128;
  float* scores = (float*)p; p += 128;
  int*   prev   = (int*)p;   p += 128;
  int*   stopped= (int*)p;   p += 64;
  int*   bufA   = (int*)p;   p += NBK * TT * 4;
  int*   bufB   = (int*)p;   p += NBK * TT * 4;

  k_init<<<(NBK * DD + 255) / 256, 256, 0, stream>>>(words, h0, c0, wbuf, scores, stopped,
                                                     h_cur, c_cur, bufA, bufB);
  for (int j = 0; j < TT; ++j) {
    int* bin  = (j & 1) ? bufB : bufA;
    int* bo_  = (j & 1) ? bufA : bufB;
    k_gates_lstm<<<DD / 16, 32, 0, stream>>>(E, Wx, Wh, bias, wbuf, h_cur, c_cur,
                                             h_tmp, c_tmp);
    k_logits<<<VV / 16, 32, 0, stream>>>(h_tmp, Wout, bout, logits);
    k_rowtopk<<<NBK, 256, 0, stream>>>(logits, lse, topv, topi);
    k_combine<<<1, 32, 0, stream>>>(topv, topi, lse, scores, wbuf, stopped, bin, bo_, prev, j);
    k_gather<<<NBK, 256, 0, stream>>>(h_tmp, c_tmp, prev, h_cur, c_cur);
  }
  // after 16 steps (even), final histories live in bufA
  k_final<<<3, 256, 0, stream>>>(bufA, scores, (float*)d_out);
}